// Sum_35897336659992
// MI455X (gfx1250) — compile-verified
//
#include <hip/hip_runtime.h>
#include <math.h>

// Problem constants (from reference): B=256, F=64, CIN=32, COUT=16, R=8
#define Bn    256
#define Fn    64
#define CIN   32
#define COUT  16
#define Rn    8

#define BROWS 32            // b-rows per block
#define ER    264           // E row stride in floats (256 + 8 pad -> conflict-free strided reads)
#define WPLN  512           // W per-r plane stride (CIN*COUT)
#define OUTR  132           // staged-output row stride (128 + 4 pad, 16B-aligned rows)

typedef float v2f __attribute__((ext_vector_type(2)));
typedef float v4f __attribute__((ext_vector_type(4)));
typedef float v8f __attribute__((ext_vector_type(8)));
typedef int   v4i __attribute__((ext_vector_type(4)));

// gfx1250 async global->LDS path (ASYNCcnt), with guarded fallback
#if defined(__HIP_DEVICE_COMPILE__) && __has_builtin(__builtin_amdgcn_global_load_async_to_lds_b128)
#define HAVE_ASYNC_LDS 1
typedef __attribute__((address_space(1))) v4i gv4i;   // global-AS int4 (builtin param 1)
typedef __attribute__((address_space(3))) v4i lv4i;   // LDS-AS int4    (builtin param 2)
#else
#define HAVE_ASYNC_LDS 0
#endif

__global__ __launch_bounds__(256) void mixture_lse_kernel(
    const float* __restrict__ ll,      // (B,F,CIN,R)
    const float* __restrict__ logits,  // (F,CIN,COUT,R)
    float* __restrict__ out)           // (B,F,COUT,R)
{
    // sE (8448 f) | sW (4096 f) | sM (256 f)  => 51,200 bytes total
    __shared__ float sE[BROWS * ER];
    __shared__ float sW[Rn * WPLN];
    __shared__ float sM[BROWS * Rn];
    float* sOut = sE;                  // alias: reused after WMMA reads complete

    const int tid  = threadIdx.x;
    const int f    = blockIdx.x >> 3;          // 64 f values
    const int bb   = (blockIdx.x & 7) * BROWS; // 8 chunks of 32 b-rows

    const v4f* ll4 = reinterpret_cast<const v4f*>(ll);

#if HAVE_ASYNC_LDS
    // ---- Phase 0: kick off async DMA of the ll slab into sE (overlaps phases 1-2) ----
    {
        #pragma unroll
        for (int i = 0; i < 8; ++i) {
            int idx4 = tid + i * 256;           // 2048 x b128
            int bl   = idx4 >> 6;               // 64 x 16B per row
            int p4   = idx4 & 63;
            const v4f* src = &ll4[((bb + bl) * Fn + f) * 64 + p4];
            __builtin_amdgcn_global_load_async_to_lds_b128(
                (gv4i*)src, (lv4i*)&sE[bl * ER + p4 * 4], /*offset=*/0, /*cpol=*/0);
        }
    }
#endif

    // ---- Phase 1: coalesced load logits[f] slab -> sW[r][cin][co] ----
    {
        const float* lg = logits + f * (CIN * COUT * Rn);   // 4096 floats
        #pragma unroll
        for (int i = 0; i < 16; ++i) {
            int idx = tid + i * 256;            // (cin*COUT + co)*R + r
            int cin = idx >> 7;
            int rem = idx & 127;
            int co  = rem >> 3;
            int r   = rem & 7;
            sW[r * WPLN + cin * COUT + co] = lg[idx];
        }
    }
    __syncthreads();

    // ---- Phase 2: softmax over cin for each (co,r) column (128 threads) ----
    if (tid < 128) {
        int co = tid >> 3, r = tid & 7;
        float* col = &sW[r * WPLN + co];
        float mx = -INFINITY;
        #pragma unroll
        for (int c = 0; c < CIN; ++c) mx = fmaxf(mx, col[c * COUT]);
        float s = 0.f;
        #pragma unroll
        for (int c = 0; c < CIN; ++c) s += __expf(col[c * COUT] - mx);
        float inv = 1.0f / s;
        #pragma unroll
        for (int c = 0; c < CIN; ++c) col[c * COUT] = __expf(col[c * COUT] - mx) * inv;
    }

#if HAVE_ASYNC_LDS
    // Each wave drains its own async transfers, then the barrier covers all waves.
  #if __has_builtin(__builtin_amdgcn_s_wait_asynccnt)
    __builtin_amdgcn_s_wait_asynccnt(0);
  #else
    asm volatile("s_wait_asynccnt 0" ::: "memory");
  #endif
#else
    // ---- Phase 3 (fallback): direct coalesced b128 load of ll slab -> sE ----
    {
        #pragma unroll
        for (int i = 0; i < 8; ++i) {
            int idx4 = tid + i * 256;           // 2048 float4s
            int bl   = idx4 >> 6;               // 64 float4 per row
            int p4   = idx4 & 63;
            v4f v = __builtin_nontemporal_load(&ll4[((bb + bl) * Fn + f) * 64 + p4]);
            *reinterpret_cast<v4f*>(&sE[bl * ER + p4 * 4]) = v;
        }
    }
#endif
    __syncthreads();

    // ---- Phase 4: per-(b,r) max over cin, then E = exp(ll - m) in place ----
    {
        int bl = tid >> 3, r = tid & 7;
        float* row = &sE[bl * ER + r];
        float mx = -INFINITY;
        #pragma unroll
        for (int c = 0; c < CIN; ++c) mx = fmaxf(mx, row[c * Rn]);
        sM[tid] = mx;                            // sM[bl*8 + r]
        #pragma unroll
        for (int c = 0; c < CIN; ++c) row[c * Rn] = __expf(row[c * Rn] - mx);
    }
    __syncthreads();

    // ---- Phase 5: WMMA jobs. 16 jobs = 2 mtiles x 8 r; wave w -> jobs {w, w+8} ----
    const int wave = tid >> 5;
    const int lane = tid & 31;
    const int n    = lane & 15;                  // M-row (for A) / N-col (for B)
    const int hi   = (lane >= 16) ? 1 : 0;
    const int koff = hi ? 2 : 0;                 // 32-bit A: V0={K0|K2}, V1={K1|K3}

    v8f acc[2];
    int mt[2], rr[2];
    #pragma unroll
    for (int q = 0; q < 2; ++q) {
        const int job   = wave + q * 8;
        const int mtile = job >> 3;
        const int r     = job & 7;
        mt[q] = mtile; rr[q] = r;

        v8f c = {0.f, 0.f, 0.f, 0.f, 0.f, 0.f, 0.f, 0.f};
        const float* Arow  = &sE[(mtile * 16 + n) * ER + r];  // A[m][k] = E[row][k*8 + r]
        const float* Bplan = &sW[r * WPLN + n];               // B[k][n] = W[r][k*16 + n]
        #pragma unroll
        for (int k4 = 0; k4 < 8; ++k4) {
            const int kb = k4 * 4 + koff;
            v2f a, b;
            a.x = Arow[kb * Rn];
            a.y = Arow[(kb + 1) * Rn];
            b.x = Bplan[kb * COUT];
            b.y = Bplan[(kb + 1) * COUT];
            c = __builtin_amdgcn_wmma_f32_16x16x4_f32(
                    /*neg_a=*/false, a, /*neg_b=*/false, b,
                    /*c_mod=*/(short)0, c, /*reuse_a=*/false, /*reuse_b=*/false);
        }
        acc[q] = c;
    }
    __syncthreads();   // all sE reads done; safe to alias sOut over sE

    // ---- Phase 6: D layout (VGPR i: lanes0-15 M=i, lanes16-31 M=i+8) -> sOut ----
    #pragma unroll
    for (int q = 0; q < 2; ++q) {
        #pragma unroll
        for (int i = 0; i < 8; ++i) {
            int row = mt[q] * 16 + i + (hi ? 8 : 0);
            sOut[row * OUTR + n * Rn + rr[q]] = sM[row * Rn + rr[q]] + __logf(acc[q][i]);
        }
    }
    __syncthreads();

    // ---- Phase 7: coalesced non-temporal b128 writeback (32 rows x 128 floats) ----
    {
        v4f* out4 = reinterpret_cast<v4f*>(out);
        #pragma unroll
        for (int i = 0; i < 4; ++i) {
            int idx4 = tid + i * 256;            // 1024 float4s
            int bl   = idx4 >> 5;                // 32 float4 per row
            int p4   = idx4 & 31;
            v4f v = *reinterpret_cast<const v4f*>(&sOut[bl * OUTR + p4 * 4]);
            __builtin_nontemporal_store(v, &out4[((bb + bl) * Fn + f) * (COUT * Rn / 4) + p4]);
        }
    }
}

extern "C" void kernel_launch(void* const* d_in, const int* in_sizes, int n_in,
                              void* d_out, int out_size, void* d_ws, size_t ws_size,
                              hipStream_t stream) {
    (void)in_sizes; (void)n_in; (void)d_ws; (void)ws_size; (void)out_size;
    const float* ll     = (const float*)d_in[0];   // (256,64,32,8) fp32
    const float* logits = (const float*)d_in[1];   // (64,32,16,8) fp32
    float*       out    = (float*)d_out;           // (256,64,16,8) fp32

    // 64 f-slices x 8 chunks of 32 batch rows = 512 blocks, 256 threads (8 waves)
    mixture_lse_kernel<<<dim3(Fn * (Bn / BROWS)), dim3(256), 0, stream>>>(ll, logits, out);
}